// MERG_25632364822803
// MI455X (gfx1250) — compile-verified
//
#include <hip/hip_runtime.h>
#include <hip/hip_bf16.h>

// ---------------- problem constants (match reference) ----------------
#define NN      2048
#define IN_DIM  128
#define HID     64
#define NE      32768
#define EPSV    1e-5f

typedef float v2f  __attribute__((ext_vector_type(2)));
typedef float v8f  __attribute__((ext_vector_type(8)));
typedef unsigned int u32x4 __attribute__((ext_vector_type(4)));
typedef unsigned int u32x8 __attribute__((ext_vector_type(8)));

// ---------------------------------------------------------------------
// small helpers
// ---------------------------------------------------------------------
__device__ __forceinline__ float wave_sum32(float v) {
    #pragma unroll
    for (int m = 16; m >= 1; m >>= 1) v += __shfl_xor(v, m, 32);
    return v;
}

// ---------------------------------------------------------------------
// bucket-building kernels (group edges by src node)
// ---------------------------------------------------------------------
__global__ void zero_ints_kernel(int* p, int n) {
    int i = blockIdx.x * blockDim.x + threadIdx.x;
    if (i < n) p[i] = 0;
}

__global__ void count_kernel(const int* __restrict__ src, int* __restrict__ counts) {
    int e = blockIdx.x * blockDim.x + threadIdx.x;
    if (e < NE) atomicAdd(&counts[src[e]], 1);
}

__global__ void scan_kernel(const int* __restrict__ counts, int* __restrict__ offs) {
    if (blockIdx.x == 0 && threadIdx.x == 0) {
        int acc = 0;
        for (int i = 0; i < NN; ++i) { offs[i] = acc; acc += counts[i]; }
        offs[NN] = acc;
    }
}

__global__ void scatter_kernel(const int* __restrict__ src, const int* __restrict__ offs,
                               int* __restrict__ cursor, int* __restrict__ perm) {
    int e = blockIdx.x * blockDim.x + threadIdx.x;
    if (e < NE) {
        int s = src[e];
        int pos = atomicAdd(&cursor[s], 1);
        perm[offs[s] + pos] = e;  // order within bucket irrelevant: per-edge output rows
    }
}

// ---------------------------------------------------------------------
// Generic fp32 WMMA GEMM: out[M,Ncols] = epilogue(A[M,K] @ B[K,Ncols])
//   mode 0: relu(acc + bias[n])
//   mode 1: relu((acc + bias[n]) * gamma[n]*inv + beta[n])              (BN eval)
//   mode 2: mode1 + add0[m,n] + add1[m,n]                               (final sum)
// One wave per 16(m) x 64(n) output strip: 4 accumulators share each
// A fragment (4 V_WMMA_F32_16X16X4_F32 per K-step, A loaded once).
// M mult of 16, Ncols mult of 64, K mult of 4.
// ---------------------------------------------------------------------
__global__ void gemm_wmma_f32_kernel(const float* __restrict__ A, const float* __restrict__ B,
                                     const float* __restrict__ bias,
                                     const float* __restrict__ gamma, const float* __restrict__ beta,
                                     const float* __restrict__ add0, const float* __restrict__ add1,
                                     float* __restrict__ out,
                                     int M, int K, int Ncols, int mode) {
    const int wave = threadIdx.x >> 5;
    const int lane = threadIdx.x & 31;
    const int strip = blockIdx.x * 8 + wave;
    const int strips_n = Ncols >> 6;               // 64-wide strips
    const int sm = strip / strips_n;
    const int sn = strip - sm * strips_n;
    if (sm >= (M >> 4)) return;                    // uniform per wave: EXEC stays all-1s

    const int half = lane >> 4;                    // K-half select (ISA fp32 A/B layout)
    const int q    = lane & 15;                    // row (A) / col (B,C,D) within tile

    const float* Arow  = A + (size_t)(sm * 16 + q) * K;
    const int    nbase = sn * 64 + q;

    v8f c0 = {}, c1 = {}, c2 = {}, c3 = {};
    #pragma unroll 4
    for (int k0 = 0; k0 < K; k0 += 4) {
        v2f a = *(const v2f*)&Arow[k0 + 2 * half];             // K = 2*half, 2*half+1
        const float* Br0 = B + (size_t)(k0 + 2 * half) * Ncols + nbase;
        const float* Br1 = Br0 + Ncols;
        v2f b0; b0.x = Br0[0];  b0.y = Br1[0];
        v2f b1; b1.x = Br0[16]; b1.y = Br1[16];
        v2f b2; b2.x = Br0[32]; b2.y = Br1[32];
        v2f b3; b3.x = Br0[48]; b3.y = Br1[48];
        c0 = __builtin_amdgcn_wmma_f32_16x16x4_f32(false, a, false, b0, (short)0, c0, false, false);
        c1 = __builtin_amdgcn_wmma_f32_16x16x4_f32(false, a, false, b1, (short)0, c1, false, false);
        c2 = __builtin_amdgcn_wmma_f32_16x16x4_f32(false, a, false, b2, (short)0, c2, false, false);
        c3 = __builtin_amdgcn_wmma_f32_16x16x4_f32(false, a, false, b3, (short)0, c3, false, false);
    }

    const float invs = rsqrtf(1.0f + EPSV);
    v8f cs[4] = { c0, c1, c2, c3 };
    #pragma unroll
    for (int sub = 0; sub < 4; ++sub) {
        int n = sn * 64 + sub * 16 + q;
        float biasn = bias[n];
        float gm = 0.0f, bt = 0.0f;
        if (mode >= 1) { gm = gamma[n] * invs; bt = beta[n]; }
        #pragma unroll
        for (int j = 0; j < 8; ++j) {
            int m = sm * 16 + j + 8 * half;        // C/D: vgpr j -> rows j / j+8
            size_t idx = (size_t)m * Ncols + n;
            float v = cs[sub][j] + biasn;
            if (mode >= 1) v = v * gm + bt;
            v = fmaxf(v, 0.0f);
            if (mode == 2) v += add0[idx] + add1[idx];
            out[idx] = v;
        }
    }
}

// ---------------------------------------------------------------------
// Local branch front-end: per edge, LayerNorm both gathered rows,
// Conv1d(2->1,k=3,pad=1) over the 128-dim axis, ReLU -> r[E,128].
// One wave (32 lanes) per edge, 8 edges per block.
// ---------------------------------------------------------------------
__global__ void local_conv_kernel(const float* __restrict__ local,
                                  const int* __restrict__ src, const int* __restrict__ dst,
                                  const float* __restrict__ ln_w, const float* __restrict__ ln_b,
                                  const float* __restrict__ conv_w, const float* __restrict__ conv_b,
                                  float* __restrict__ r) {
    __shared__ float rows[8][2][132];                      // +2 zero padding, +2 align slop
    const int wave = threadIdx.x >> 5;
    const int lane = threadIdx.x & 31;
    const int e = blockIdx.x * 8 + wave;

    const int nodes[2] = { src[e], dst[e] };
    const float w00 = conv_w[0], w01 = conv_w[1], w02 = conv_w[2];
    const float w10 = conv_w[3], w11 = conv_w[4], w12 = conv_w[5];
    const float cb = conv_b[0];

    #pragma unroll
    for (int rr = 0; rr < 2; ++rr) {
        const float* x = local + (size_t)nodes[rr] * IN_DIM;
        float v[4], s = 0.0f, s2 = 0.0f;
        #pragma unroll
        for (int j = 0; j < 4; ++j) {
            v[j] = x[lane + 32 * j];
            s += v[j]; s2 += v[j] * v[j];
        }
        s  = wave_sum32(s);
        s2 = wave_sum32(s2);
        const float mu   = s * (1.0f / IN_DIM);
        const float var  = s2 * (1.0f / IN_DIM) - mu * mu; // biased var (jnp.var)
        const float rstd = rsqrtf(var + EPSV);
        #pragma unroll
        for (int j = 0; j < 4; ++j) {
            int i = lane + 32 * j;
            rows[wave][rr][1 + i] = (v[j] - mu) * rstd * ln_w[i] + ln_b[i];
        }
        if (lane == 0) { rows[wave][rr][0] = 0.0f; rows[wave][rr][129] = 0.0f; }
    }
    __syncthreads();

    #pragma unroll
    for (int j = 0; j < 4; ++j) {
        int i = lane + 32 * j;
        float y = w00 * rows[wave][0][i] + w01 * rows[wave][0][i + 1] + w02 * rows[wave][0][i + 2]
                + w10 * rows[wave][1][i] + w11 * rows[wave][1][i + 1] + w12 * rows[wave][1][i + 2]
                + cb;
        r[(size_t)e * IN_DIM + i] = fmaxf(y, 0.0f);
    }
}

// ---------------------------------------------------------------------
// Global-branch einsum as grouped WMMA GEMM.
// One block per src node i: TDM (tensor_load_to_lds) DMAs the 16 KB
// p1[i] tile into LDS with 2-DWORD padding every 64 DWORDs (stride-66
// rows -> no bank conflicts), tracked by TENSORcnt. Then for each
// 16-edge panel of the bucket, gather p2[dst] columns into LDS and run
// 16x16 WMMA tiles: [64(d) x 16(e)] = P1_i[64x64] @ P2cols[64x16].
// Epilogue applies BN(bng): Z[e,d] = pre*bng_g[d]*inv + bng_b[d].
// ---------------------------------------------------------------------
__global__ void edge_gemm_kernel(const float* __restrict__ p1, const float* __restrict__ p2,
                                 const int* __restrict__ dst,
                                 const int* __restrict__ perm, const int* __restrict__ offs,
                                 const float* __restrict__ bng_g, const float* __restrict__ bng_b,
                                 float* __restrict__ Z) {
    __shared__ float P1s[HID * 66];        // row stride 66 (TDM pad) -> conflict-free
    __shared__ float Bc[2][HID * 16];      // [panel][k*16 + col]

    const int i = blockIdx.x;              // src node
    const int t = threadIdx.x;

    // ---- TDM: async DMA of p1 row (4096 f32, contiguous) into padded LDS ----
    if (t < 32) {                          // one wave issues; EXEC ignored by TDM
        unsigned int lds_off = (unsigned int)(size_t)(void*)&P1s[0];   // flat low 32 = LDS byte addr
        unsigned long long ga = (unsigned long long)(size_t)(p1 + (size_t)i * (HID * HID));
        u32x4 g0;
        g0[0] = 1u;                                        // count=1 (valid), user mode
        g0[1] = lds_off;                                   // lds_addr
        g0[2] = (unsigned int)ga;                          // global_addr[31:0]
        g0[3] = (unsigned int)((ga >> 32) & 0x01FFFFFFu)   // global_addr[56:32]
              | 0x80000000u;                               // type=2 ("image")
        u32x8 g1;
        g1[0] = (2u << 16)                                 // data_size = 4B
              | (1u << 20)                                 // pad_enable
              | (5u << 22)                                 // pad_interval: 64 DWORDs
              | (1u << 25);                                // pad_amount: 2 DWORDs
        g1[1] = (4096u & 0xFFFFu) << 16;                   // tensor_dim0 = 4096 (lo16)
        g1[2] = (4096u >> 16) | (1u << 16);                // tensor_dim0 hi | tensor_dim1 = 1 (lo16)
        g1[3] = (4096u << 16);                             // tensor_dim1 hi=0 | tile_dim0 = 4096
        g1[4] = 1u;                                        // tile_dim1 = 1, tile_dim2 = 0
        g1[5] = 4096u;                                     // tensor_dim0_stride lo32
        g1[6] = 0u;                                        // stride hi | dim1_stride lo
        g1[7] = 0u;
        asm volatile("tensor_load_to_lds %0, %1" :: "s"(g0), "s"(g1) : "memory");
        __builtin_amdgcn_s_wait_tensorcnt(0);
    }
    __syncthreads();                       // P1s visible to all waves

    const int start = offs[i];
    const int count = offs[i + 1] - start;

    const int wave = t >> 5, lane = t & 31;
    const int half = lane >> 4, q = lane & 15;
    const int panel = wave >> 2;           // 0/1: which 16-edge panel
    const int dt    = wave & 3;            // which 16-row d tile
    const float invs = rsqrtf(1.0f + EPSV);

    for (int base = 0; base < count; base += 32) {
        __syncthreads();                   // Bc free from previous iteration
        {   // gather 2 panels x 16 cols x 64 k = 2048 floats; 8 per thread
            int colg = t >> 3;             // 0..31
            int kb   = (t & 7) * 8;        // k block
            int ch = colg >> 4, col = colg & 15;
            int eidx = base + ch * 16 + col;
            if (eidx < count) {
                int eg = perm[start + eidx];
                const float* p2r = p2 + (size_t)dst[eg] * HID;
                #pragma unroll
                for (int kk = 0; kk < 8; ++kk) Bc[ch][(kb + kk) * 16 + col] = p2r[kb + kk];
            } else {
                #pragma unroll
                for (int kk = 0; kk < 8; ++kk) Bc[ch][(kb + kk) * 16 + col] = 0.0f;
            }
        }
        __syncthreads();

        v8f c = {};
        #pragma unroll 4
        for (int k0 = 0; k0 < HID; k0 += 4) {
            v2f a = *(const v2f*)&P1s[(dt * 16 + q) * 66 + k0 + 2 * half];
            v2f b;
            b.x = Bc[panel][(k0 + 2 * half)     * 16 + q];
            b.y = Bc[panel][(k0 + 2 * half + 1) * 16 + q];
            c = __builtin_amdgcn_wmma_f32_16x16x4_f32(false, a, false, b, (short)0, c, false, false);
        }

        #pragma unroll
        for (int j = 0; j < 8; ++j) {
            int d = dt * 16 + j + 8 * half;
            int eidx = base + panel * 16 + q;
            if (eidx < count) {
                int eg = perm[start + eidx];
                Z[(size_t)eg * HID + d] = c[j] * (bng_g[d] * invs) + bng_b[d];
            }
        }
    }
}

// ---------------------------------------------------------------------
// launcher
// ---------------------------------------------------------------------
extern "C" void kernel_launch(void* const* d_in, const int* in_sizes, int n_in,
                              void* d_out, int out_size, void* d_ws, size_t ws_size,
                              hipStream_t stream) {
    const float* h      = (const float*)d_in[0];
    const float* local  = (const float*)d_in[1];
    const float* e_in   = (const float*)d_in[2];
    const int*   src    = (const int*)d_in[3];
    const int*   dst    = (const int*)d_in[4];
    const float* ln_w   = (const float*)d_in[5];
    const float* ln_b   = (const float*)d_in[6];
    const float* conv_w = (const float*)d_in[7];
    const float* conv_b = (const float*)d_in[8];
    const float* W1     = (const float*)d_in[9];
    const float* b1     = (const float*)d_in[10];
    const float* Wp2    = (const float*)d_in[11];
    const float* bp2    = (const float*)d_in[12];
    const float* W2     = (const float*)d_in[13];
    const float* b2     = (const float*)d_in[14];
    const float* W3     = (const float*)d_in[15];
    const float* b3     = (const float*)d_in[16];
    const float* bng_g  = (const float*)d_in[17];
    const float* bng_b  = (const float*)d_in[18];
    const float* bnG_g  = (const float*)d_in[19];
    const float* bnG_b  = (const float*)d_in[20];
    const float* bnL_g  = (const float*)d_in[21];
    const float* bnL_b  = (const float*)d_in[22];

    // workspace carve-up (~67 MB)
    char* ws = (char*)d_ws;
    float* p1 = (float*)ws;                 ws += (size_t)NN * HID * HID * sizeof(float); // 32 MB
    float* p2 = (float*)ws;                 ws += (size_t)NN * HID * sizeof(float);       // 0.5 MB
    float* rL = (float*)ws;                 ws += (size_t)NE * IN_DIM * sizeof(float);    // 16 MB
    float* Lc = (float*)ws;                 ws += (size_t)NE * HID * sizeof(float);       // 8 MB
    float* Z  = (float*)ws;                 ws += (size_t)NE * HID * sizeof(float);       // 8 MB
    int* counts = (int*)ws;                 ws += (size_t)NN * sizeof(int);
    int* cursor = (int*)ws;                 ws += (size_t)NN * sizeof(int);
    int* offs   = (int*)ws;                 ws += (size_t)(NN + 1) * sizeof(int);
    int* perm   = (int*)ws;                 ws += (size_t)NE * sizeof(int);

    // 1) bucket edges by src
    zero_ints_kernel<<<(2 * NN + 255) / 256, 256, 0, stream>>>(counts, 2 * NN); // counts+cursor contiguous
    count_kernel<<<NE / 256, 256, 0, stream>>>(src, counts);
    scan_kernel<<<1, 64, 0, stream>>>(counts, offs);
    scatter_kernel<<<NE / 256, 256, 0, stream>>>(src, offs, cursor, perm);

    // 2) p1 = relu(h @ W1 + b1)   [2048 x 4096], K=128 -> 8192 strips / 8 waves
    gemm_wmma_f32_kernel<<<1024, 256, 0, stream>>>(h, W1, b1, nullptr, nullptr, nullptr, nullptr,
                                                   p1, NN, IN_DIM, HID * HID, 0);
    // 3) p2 = relu(h @ Wp2 + bp2) [2048 x 64] -> 128 strips
    gemm_wmma_f32_kernel<<<16, 256, 0, stream>>>(h, Wp2, bp2, nullptr, nullptr, nullptr, nullptr,
                                                 p2, NN, IN_DIM, HID, 0);

    // 4) local branch: LN + conv + relu -> rL[E,128]
    local_conv_kernel<<<NE / 8, 256, 0, stream>>>(local, src, dst, ln_w, ln_b, conv_w, conv_b, rL);
    // 5) Lc = relu(bnL(rL @ W2 + b2))  [32768 x 64], K=128 -> 2048 strips
    gemm_wmma_f32_kernel<<<256, 256, 0, stream>>>(rL, W2, b2, bnL_g, bnL_b, nullptr, nullptr,
                                                  Lc, NE, IN_DIM, HID, 1);

    // 6) grouped einsum + bn(bng): Z[e,d]
    edge_gemm_kernel<<<NN, 256, 0, stream>>>(p1, p2, dst, perm, offs, bng_g, bng_b, Z);

    // 7) out = relu(bnG(Z @ W3 + b3)) + Lc + e   [32768 x 64], K=64 -> 2048 strips
    gemm_wmma_f32_kernel<<<256, 256, 0, stream>>>(Z, W3, b3, bnG_g, bnG_b, Lc, e_in,
                                                  (float*)d_out, NE, HID, HID, 2);
}